// OpticalFlowLK_20856361189457
// MI455X (gfx1250) — compile-verified
//
#include <hip/hip_runtime.h>
#include <stdint.h>

#ifndef __has_builtin
#define __has_builtin(x) 0
#endif

#define H_DIM 2048
#define W_DIM 2048
#define TILE  32

// Image LDS tile: 35 rows x 40 cols (col window starts at c0-4 for 16B-aligned
// b128 async loads; the stencil only reads local cols 3..37 <-> gcol c0-1..c0+33).
#define IMG_R 35
#define IMG_STRIDE 40        // dwords; 160 B row stride keeps 16B chunk alignment
// Derivative tile: 34x34 (one ring around the 32x32 output tile)
#define RES_R 34
#define RES_C 34
#define RES_STRIDE 36

typedef int v4i __attribute__((ext_vector_type(4)));

// ---- CDNA5 async global->LDS copies (ASYNCcnt path) -----------------------
__device__ __forceinline__ void async_g2l_b32(const float* gsrc, const float* ldsdst) {
  // Low 32 bits of a flat LDS address are the LDS byte offset (ISA 10.2).
  uint32_t lds_off = (uint32_t)(uintptr_t)ldsdst;
#if __has_builtin(__builtin_amdgcn_global_load_async_to_lds_b32)
  typedef __attribute__((address_space(1))) int GI;
  typedef __attribute__((address_space(3))) int LI;
  __builtin_amdgcn_global_load_async_to_lds_b32(
      (GI*)(uintptr_t)gsrc, (LI*)(uintptr_t)lds_off, /*offset=*/0, /*cpol=*/0);
#else
  asm volatile("global_load_async_to_lds_b32 %0, %1, off"
               :: "v"(lds_off), "v"((uint64_t)(uintptr_t)gsrc)
               : "memory");
#endif
}

__device__ __forceinline__ void async_g2l_b128(const float* gsrc, const float* ldsdst) {
  uint32_t lds_off = (uint32_t)(uintptr_t)ldsdst;
#if __has_builtin(__builtin_amdgcn_global_load_async_to_lds_b128)
  typedef __attribute__((address_space(1))) v4i GV;
  typedef __attribute__((address_space(3))) v4i LV;
  __builtin_amdgcn_global_load_async_to_lds_b128(
      (GV*)(uintptr_t)gsrc, (LV*)(uintptr_t)lds_off, /*offset=*/0, /*cpol=*/0);
#else
  asm volatile("global_load_async_to_lds_b128 %0, %1, off"
               :: "v"(lds_off), "v"((uint64_t)(uintptr_t)gsrc)
               : "memory");
#endif
}

__device__ __forceinline__ void wait_async0() {
#if __has_builtin(__builtin_amdgcn_s_wait_asynccnt)
  __builtin_amdgcn_s_wait_asynccnt(0);
#else
  asm volatile("s_wait_asynccnt 0" ::: "memory");
#endif
}

// ---------------------------------------------------------------------------
__global__ __launch_bounds__(256)
void OpticalFlowLK_kernel(const float* __restrict__ prev,
                          const float* __restrict__ next,
                          float* __restrict__ out) {
  __shared__ __align__(16) float sp[IMG_R * IMG_STRIDE];
  __shared__ __align__(16) float sn[IMG_R * IMG_STRIDE];
  __shared__ float rxs[RES_R * RES_STRIDE];
  __shared__ float rys[RES_R * RES_STRIDE];
  __shared__ float rts[RES_R * RES_STRIDE];

  const int tid = threadIdx.y * 32 + threadIdx.x;
  const int r0  = blockIdx.y * TILE;
  const int c0  = blockIdx.x * TILE;

  // ---- Stage 1: async-copy halo tiles of both images into LDS -------------
  // Row mapping handles reflect pad (padded row H -> H-2); rows feeding only
  // masked-out derivative rows are clamped (value irrelevant, address legal).
  const bool fastcols = (c0 >= 4) && (c0 + 35 < W_DIM);   // full 40-col window in-bounds
  if (fastcols) {
    // 10 x b128 chunks per row per image; global col base c0-4 is 16B aligned.
    for (int k = tid; k < IMG_R * 10; k += 256) {
      int i  = k / 10;
      int ch = k - i * 10;
      int gr = r0 - 1 + i;
      int lr = (gr == H_DIM) ? (H_DIM - 2) : (gr < 0 ? 0 : (gr > H_DIM - 1 ? H_DIM - 1 : gr));
      size_t g = (size_t)lr * W_DIM + (size_t)(c0 - 4 + ch * 4);
      int    l = i * IMG_STRIDE + ch * 4;
      async_g2l_b128(prev + g, &sp[l]);
      async_g2l_b128(next + g, &sn[l]);
    }
  } else {
    // Edge block-columns: per-element b32 with per-column reflect/clamp.
    for (int k = tid; k < IMG_R * 35; k += 256) {
      int i  = k / 35;
      int j  = k - i * 35;                 // local col m = j+3, gcol = c0-1+j
      int gr = r0 - 1 + i;
      int gc = c0 - 1 + j;
      int lr = (gr == H_DIM) ? (H_DIM - 2) : (gr < 0 ? 0 : (gr > H_DIM - 1 ? H_DIM - 1 : gr));
      int lc = (gc == W_DIM) ? (W_DIM - 2) : (gc < 0 ? 0 : (gc > W_DIM - 1 ? W_DIM - 1 : gc));
      size_t g = (size_t)lr * W_DIM + (size_t)lc;
      int    l = i * IMG_STRIDE + (j + 3);
      async_g2l_b32(prev + g, &sp[l]);
      async_g2l_b32(next + g, &sn[l]);
    }
  }
  wait_async0();
  __syncthreads();

  // ---- Stage 2: derivative tiles rx, ry, rt (34x34), zero outside image ---
  for (int k = tid; k < RES_R * RES_C; k += 256) {
    int i  = k / RES_C;
    int j  = k - i * RES_C;
    int gr = r0 - 1 + i;
    int gc = c0 - 1 + j;
    float rx = 0.f, ry = 0.f, rt = 0.f;
    if (gr >= 0 && gr < H_DIM && gc >= 0 && gc < W_DIM) {
      int b = i * IMG_STRIDE + (j + 3);    // image local col = gcol - (c0-4)
      float p00 = sp[b],              p01 = sp[b + 1];
      float p10 = sp[b + IMG_STRIDE], p11 = sp[b + IMG_STRIDE + 1];
      float n00 = sn[b],              n01 = sn[b + 1];
      float n10 = sn[b + IMG_STRIDE], n11 = sn[b + IMG_STRIDE + 1];
      float gxp = -p00 + p01 - p10 + p11;
      float gyp = -p00 - p01 + p10 + p11;
      float gsp =  p00 + p01 + p10 + p11;
      float gxn = -n00 + n01 - n10 + n11;
      float gyn = -n00 - n01 + n10 + n11;
      float gsn =  n00 + n01 + n10 + n11;
      rx = 0.5f * (gxp + gxn);
      ry = 0.5f * (gyp + gyn);
      rt = 0.5f * (gsn - gsp);
    }
    int o = i * RES_STRIDE + j;
    rxs[o] = rx; rys[o] = ry; rts[o] = rt;
  }
  __syncthreads();

  // ---- Stage 3: 3x3 box sums of the five products + 2x2 solve -------------
  const int ox = threadIdx.x;
  #pragma unroll
  for (int q = 0; q < 4; ++q) {
    int oy = (int)threadIdx.y + 8 * q;
    float Sxx = 0.f, Syy = 0.f, Sxy = 0.f, Sxt = 0.f, Syt = 0.f;
    #pragma unroll
    for (int di = 0; di < 3; ++di) {
      int base = (oy + di) * RES_STRIDE + ox;
      #pragma unroll
      for (int dj = 0; dj < 3; ++dj) {
        float rx = rxs[base + dj];
        float ry = rys[base + dj];
        float rt = rts[base + dj];
        Sxx += rx * rx;
        Syy += ry * ry;
        Sxy += rx * ry;
        Sxt += rt * rx;
        Syt += rt * ry;
      }
    }
    float det = Sxx * Syy - Sxy * Sxy;
    int r = r0 + oy;
    int c = c0 + ox;
    float u = 0.f, v = 0.f;
    if (det != 0.f && r != 0 && c != 0) {
      u = (Syy * Sxt - Sxy * Syt) / det;
      v = (Sxx * Syt - Sxy * Sxt) / det;
    }
    size_t idx = (size_t)r * W_DIM + (size_t)c;
    out[idx] = u;
    out[(size_t)H_DIM * W_DIM + idx] = v;
  }
}

extern "C" void kernel_launch(void* const* d_in, const int* in_sizes, int n_in,
                              void* d_out, int out_size, void* d_ws, size_t ws_size,
                              hipStream_t stream) {
  (void)in_sizes; (void)n_in; (void)out_size; (void)d_ws; (void)ws_size;
  const float* prev = (const float*)d_in[0];
  const float* next = (const float*)d_in[1];
  float* out = (float*)d_out;
  dim3 block(32, 8);                       // 8 wave32s per block
  dim3 grid(W_DIM / TILE, H_DIM / TILE);   // 64 x 64 tiles
  OpticalFlowLK_kernel<<<grid, block, 0, stream>>>(prev, next, out);
}